// GCN_3504693313862
// MI455X (gfx1250) — compile-verified
//
#include <hip/hip_runtime.h>

typedef float v2f __attribute__((ext_vector_type(2)));
typedef float v8f __attribute__((ext_vector_type(8)));

#define FEATS 64
#define HID   64
#define NCLS  20

// ---------------------------------------------------------------------------
// Zero a float buffer with float4 stores.
// ---------------------------------------------------------------------------
__global__ void gcn_zero4(float4* __restrict__ p, int n4) {
    int i = blockIdx.x * blockDim.x + threadIdx.x;
    if (i < n4) p[i] = make_float4(0.f, 0.f, 0.f, 0.f);
}

// ---------------------------------------------------------------------------
// Edge scatter: agg[dst] += feat[src], 64 feats per edge.
// One thread handles one (edge, 4-feature group): float4 gather + 4 atomics.
// Index loads are 16x redundant per edge but hit the same cacheline in-wave.
// ---------------------------------------------------------------------------
__global__ void gcn_scatter64(const float* __restrict__ feat,
                              const int*   __restrict__ src,
                              const int*   __restrict__ dst,
                              float*       __restrict__ agg,
                              int nedges) {
    int gid = blockIdx.x * blockDim.x + threadIdx.x;
    int e   = gid >> 4;
    if (e >= nedges) return;
    int fo  = (gid & 15) * 4;
    int s   = src[e];
    int d   = dst[e];
    const float4 v = *(const float4*)(feat + s * FEATS + fo);
    float* p = agg + d * FEATS + fo;
    atomicAdd(p + 0, v.x);
    atomicAdd(p + 1, v.y);
    atomicAdd(p + 2, v.z);
    atomicAdd(p + 3, v.w);
}

// ---------------------------------------------------------------------------
// GEMM1 + bias + ReLU: h[N x 64] = relu(agg[N x 64] @ W1[64 x 64] + b1)
// fp32 WMMA 16x16x4. One wave owns a 16-row stripe, all 64 output columns.
//
// A 16x4 fp32 layout (ISA 7.12.2): lanes 0-15 -> M=lane, K=k0+{0,1};
//                                  lanes 16-31 -> M=lane-16, K=k0+{2,3}.
// B 4x16 fp32 layout (symmetric):  lanes 0-15 -> N=lane, K=k0+{0,1};
//                                  lanes 16-31 -> N=lane-16, K=k0+{2,3}.
// C/D 16x16 fp32: VGPR i -> M=i (lanes 0-15) / M=i+8 (lanes 16-31), N=lane&15.
//
// W1 is staged into LDS K-pair packed: sW[(k/2)*64 + n] = {W[k][n], W[k+1][n]}
// so each B operand is ONE ds_load_b64 into an adjacent (even) VGPR pair.
// ---------------------------------------------------------------------------
__global__ void __launch_bounds__(256)
gcn_gemm1_relu(const float* __restrict__ agg,
               const float* __restrict__ W1,
               const float* __restrict__ b1,
               float*       __restrict__ h,
               int nnodes) {
    __shared__ v2f sW[(HID / 2) * HID];        // 16 KB, K-pair packed
    {
        float* sWf = (float*)sW;
        for (int i = threadIdx.x; i < HID * HID; i += 256) {
            int k = i / HID, n = i % HID;
            sWf[((k >> 1) * HID + n) * 2 + (k & 1)] = W1[i];
        }
    }
    __syncthreads();

    const int wave = threadIdx.x >> 5;
    const int lane = threadIdx.x & 31;
    const int mblk = blockIdx.x * 8 + wave;
    if (mblk * 16 >= nnodes) return;           // wave-uniform: EXEC stays full

    const int hi   = lane >> 4;                // 0: low half, 1: high half
    const int col  = lane & 15;
    int row = mblk * 16 + (lane & 15);
    if (row >= nnodes) row = nnodes - 1;       // per-lane clamp (load safety)

    v8f acc[4] = {};
    #pragma unroll
    for (int k0 = 0; k0 < FEATS; k0 += 4) {
        const int ka = k0 + 2 * hi;            // even
        v2f a = *(const v2f*)(agg + row * FEATS + ka);
        #pragma unroll
        for (int nt = 0; nt < 4; ++nt) {
            v2f b = sW[(ka >> 1) * HID + nt * 16 + col];
            acc[nt] = __builtin_amdgcn_wmma_f32_16x16x4_f32(
                false, a, false, b, (short)0, acc[nt], false, false);
        }
    }

    #pragma unroll
    for (int nt = 0; nt < 4; ++nt) {
        const int c = nt * 16 + col;
        const float bias = b1[c];
        #pragma unroll
        for (int i = 0; i < 8; ++i) {
            const int r = mblk * 16 + i + 8 * hi;
            if (r < nnodes) {
                float v = acc[nt][i] + bias;
                h[r * HID + c] = v > 0.f ? v : 0.f;
            }
        }
    }
}

// ---------------------------------------------------------------------------
// GEMM2 + bias: out[N x 20] = agg[N x 64] @ W2[64 x 20] + b2
// W2 zero-padded to 64x32 and K-pair packed in LDS; stores guarded to 20 cols.
// ---------------------------------------------------------------------------
__global__ void __launch_bounds__(256)
gcn_gemm2(const float* __restrict__ agg,
          const float* __restrict__ W2,
          const float* __restrict__ b2,
          float*       __restrict__ out,
          int nnodes) {
    __shared__ v2f sW[(HID / 2) * 32];         // 8 KB, padded + K-pair packed
    {
        float* sWf = (float*)sW;
        for (int i = threadIdx.x; i < HID * 32; i += 256) {
            int k = i >> 5, c = i & 31;
            sWf[((k >> 1) * 32 + c) * 2 + (k & 1)] =
                (c < NCLS) ? W2[k * NCLS + c] : 0.f;
        }
    }
    __syncthreads();

    const int wave = threadIdx.x >> 5;
    const int lane = threadIdx.x & 31;
    const int mblk = blockIdx.x * 8 + wave;
    if (mblk * 16 >= nnodes) return;

    const int hi  = lane >> 4;
    const int col = lane & 15;
    int row = mblk * 16 + (lane & 15);
    if (row >= nnodes) row = nnodes - 1;

    v8f acc[2] = {};
    #pragma unroll
    for (int k0 = 0; k0 < HID; k0 += 4) {
        const int ka = k0 + 2 * hi;
        v2f a = *(const v2f*)(agg + row * HID + ka);
        #pragma unroll
        for (int nt = 0; nt < 2; ++nt) {
            v2f b = sW[(ka >> 1) * 32 + nt * 16 + col];
            acc[nt] = __builtin_amdgcn_wmma_f32_16x16x4_f32(
                false, a, false, b, (short)0, acc[nt], false, false);
        }
    }

    #pragma unroll
    for (int nt = 0; nt < 2; ++nt) {
        const int c = nt * 16 + col;
        if (c < NCLS) {
            const float bias = b2[c];
            #pragma unroll
            for (int i = 0; i < 8; ++i) {
                const int r = mblk * 16 + i + 8 * hi;
                if (r < nnodes) out[r * NCLS + c] = acc[nt][i] + bias;
            }
        }
    }
}

// ---------------------------------------------------------------------------
// Launch: zero agg -> scatter(x) -> gemm1+relu -> zero agg -> scatter(h) -> gemm2
// ws layout: [agg: nnodes*64 floats][h: nnodes*64 floats]
// ---------------------------------------------------------------------------
extern "C" void kernel_launch(void* const* d_in, const int* in_sizes, int n_in,
                              void* d_out, int out_size, void* d_ws, size_t ws_size,
                              hipStream_t stream) {
    const float* x   = (const float*)d_in[0];
    const int*   src = (const int*)d_in[1];
    const int*   dst = (const int*)d_in[2];
    const float* W1  = (const float*)d_in[3];
    const float* b1  = (const float*)d_in[4];
    const float* W2  = (const float*)d_in[5];
    const float* b2  = (const float*)d_in[6];
    float* out = (float*)d_out;

    const int nnodes = in_sizes[0] / FEATS;
    const int nedges = in_sizes[1];

    float* agg = (float*)d_ws;
    float* h   = agg + (size_t)nnodes * HID;

    const int naggf  = nnodes * FEATS;
    const int zgrid  = (naggf / 4 + 255) / 256;
    const int sgrid  = (nedges * 16 + 255) / 256;
    const int mblks  = (nnodes + 15) / 16;
    const int ggrid  = (mblks + 7) / 8;

    // Layer 1
    gcn_zero4<<<zgrid, 256, 0, stream>>>((float4*)agg, naggf / 4);
    gcn_scatter64<<<sgrid, 256, 0, stream>>>(x, src, dst, agg, nedges);
    gcn_gemm1_relu<<<ggrid, 256, 0, stream>>>(agg, W1, b1, h, nnodes);

    // Layer 2 (reuse agg buffer)
    gcn_zero4<<<zgrid, 256, 0, stream>>>((float4*)agg, naggf / 4);
    gcn_scatter64<<<sgrid, 256, 0, stream>>>(h, src, dst, agg, nedges);
    gcn_gemm2<<<ggrid, 256, 0, stream>>>(agg, W2, b2, out, nnodes);
}